// FeedForwardQuantum_65481071409583
// MI455X (gfx1250) — compile-verified
//
#include <hip/hip_runtime.h>
#include <math.h>

typedef __attribute__((ext_vector_type(2))) float v2f;
typedef __attribute__((ext_vector_type(4))) float v4f;
typedef __attribute__((ext_vector_type(8))) float v8f;
typedef unsigned int u32x4 __attribute__((ext_vector_type(4)));
typedef int          i32x4 __attribute__((ext_vector_type(4)));
typedef int          i32x8 __attribute__((ext_vector_type(8)));

#define TOKENS 32768
#define DMODEL 1024
#define QDIM   16
#define WAVES_PER_BLOCK  8
#define TOKENS_PER_BLOCK (WAVES_PER_BLOCK * 16)

// LDS layout (floats): sW1[1024*16] | sW2[16*1024] | sZ[8 waves * 256]
#define SW1_OFF 0
#define SW2_OFF (DMODEL * QDIM)
#define SZ_OFF  (2 * DMODEL * QDIM)
#define SMEM_FLOATS (2 * DMODEL * QDIM + WAVES_PER_BLOCK * 256)

#if defined(__has_builtin)
#  if __has_builtin(__builtin_amdgcn_tensor_load_to_lds)
#    define HAVE_TDM 1
#  endif
#endif
// amdgpu-toolchain (clang-23 + therock headers) uses the 6-arg builtin and is
// the only toolchain shipping this header; ROCm 7.2 uses the 5-arg form.
#if defined(__has_include)
#  if __has_include(<hip/amd_detail/amd_gfx1250_TDM.h>)
#    define TDM_SIX_ARGS 1
#  endif
#endif

#ifdef HAVE_TDM
// Linear global->LDS DMA of `nelem` DWORDs via the Tensor Data Mover.
// D# per CDNA5 ISA:
//  g0: [1:0]=count=1, [63:32]=lds_addr, [120:64]=global_addr, [127:126]=type=2
//  g1: [17:16]=data_size=2 (4B), tensor_dim0=tile_dim0=nelem, dim0_stride=nelem
//  g2/g3: zero (tile <= 2D)
__device__ __forceinline__ void tdm_copy_dwords(const void* gptr,
                                                unsigned lds_byte_off,
                                                unsigned nelem) {
    unsigned long long ga = (unsigned long long)(uintptr_t)gptr;
    u32x4 g0;
    g0[0] = 1u;                                           // count=1, user D#
    g0[1] = lds_byte_off;                                 // lds_addr (bytes)
    g0[2] = (unsigned)(ga & 0xFFFFFFFFu);                 // global_addr[31:0]
    g0[3] = (unsigned)((ga >> 32) & 0x1FFFFFFu) | (2u << 30); // [56:32] | type=2
    i32x8 g1;
    g1[0] = (int)(2u << 16);                              // data_size = 4 bytes
    g1[1] = (int)((nelem & 0xFFFFu) << 16);               // tensor_dim0[15:0]
    g1[2] = (int)((nelem >> 16) & 0xFFFFu);               // tensor_dim0[31:16]
    g1[3] = (int)((nelem & 0xFFFFu) << 16);               // tile_dim0
    g1[4] = 0;                                            // tile_dim1/2 unused
    g1[5] = (int)nelem;                                   // tensor_dim0_stride
    g1[6] = 0;
    g1[7] = 0;
    i32x4 z4 = {0, 0, 0, 0};
#ifdef TDM_SIX_ARGS
    i32x8 z8 = {0, 0, 0, 0, 0, 0, 0, 0};
    __builtin_amdgcn_tensor_load_to_lds(g0, g1, z4, z4, z8, 0);
#else
    __builtin_amdgcn_tensor_load_to_lds(g0, g1, z4, z4, 0);
#endif
}
#endif

__global__ void __launch_bounds__(256)
ffq_fused_wmma(const float* __restrict__ x,
               const float* __restrict__ W1,
               const float* __restrict__ b1,
               const float* __restrict__ theta,
               const float* __restrict__ W2,
               const float* __restrict__ b2,
               float* __restrict__ out)
{
    extern __shared__ float smem[];
    float* sW1 = smem + SW1_OFF;
    float* sW2 = smem + SW2_OFF;
    float* sZ  = smem + SZ_OFF;

    const int tid  = threadIdx.x;
    const int lane = tid & 31;
    const int wave = tid >> 5;

    // ---- Stage W1 (DxQ) and W2 (QxD) into LDS (128 KB, once per block)
#ifdef HAVE_TDM
    if (wave == 0) {   // TDM is a wave-level DMA; issue from one wave only
        tdm_copy_dwords(W1, SW1_OFF * 4u, DMODEL * QDIM);
        tdm_copy_dwords(W2, SW2_OFF * 4u, DMODEL * QDIM);
    }
    __builtin_amdgcn_s_wait_tensorcnt(0);
#else
    {
        const v4f* g1 = (const v4f*)W1;
        const v4f* g2 = (const v4f*)W2;
        v4f* s1 = (v4f*)sW1;
        v4f* s2 = (v4f*)sW2;
        #pragma unroll
        for (int i = 0; i < (DMODEL * QDIM / 4) / 256; ++i) {
            s1[tid + i * 256] = g1[tid + i * 256];
            s2[tid + i * 256] = g2[tid + i * 256];
        }
    }
#endif
    __syncthreads();

    const int tileBase = blockIdx.x * TOKENS_PER_BLOCK + wave * 16;
    const int row   = lane & 15;          // M for A-frags, N(=q) for B/C frags
    const int khalf = (lane >> 4) << 1;   // 0 for lanes 0-15, 2 for lanes 16-31
    const int q     = row;

    // ---- GEMM1: h[16x16] = x_tile[16x1024] * W1[1024x16]  (fp32 WMMA, K=4/step)
    // A layout (16x4 f32): VGPR0 <- K = khalf+0, VGPR1 <- K = khalf+1
    v8f acc = {};
    const float* xrow = x + (size_t)(tileBase + row) * DMODEL;
    for (int k = 0; k < DMODEL; k += 4) {
        const int kk = k + khalf;
        v2f a = *(const v2f*)(xrow + kk);            // global_load_b64
        v2f b;
        b.x = sW1[kk * QDIM + q];                    // B row kk,   col q
        b.y = sW1[(kk + 1) * QDIM + q];              // B row kk+1, col q
        acc = __builtin_amdgcn_wmma_f32_16x16x4_f32(
            false, a, false, b, (short)0, acc, false, false);
    }

    // ---- Epilogue: z = cos(relu(h + b1) + theta); spill 16x16 z to LDS
    const float bias1 = b1[q];
    const float th    = theta[q];
    float* zbuf = sZ + wave * 256;
    #pragma unroll
    for (int v = 0; v < 8; ++v) {
        const int t = v + ((lane >> 4) << 3);        // C layout: token row
        float h = fmaxf(acc[v] + bias1, 0.0f);
        zbuf[t * 16 + q] = cosf(h + th);
    }
    __asm__ volatile("s_wait_dscnt 0x0" ::: "memory");

    // Re-read z as A-fragments (hoisted; reused for all 64 N-tiles of D)
    v2f az[4];
    #pragma unroll
    for (int c = 0; c < 4; ++c) {
        const int kk = 4 * c + khalf;
        az[c].x = zbuf[row * 16 + kk];
        az[c].y = zbuf[row * 16 + kk + 1];
    }

    // ---- GEMM2: out_tile[16x1024] = z[16x16] * W2[16x1024] + b2
    for (int n = 0; n < DMODEL / 16; ++n) {
        v8f o = {};
        #pragma unroll
        for (int c = 0; c < 4; ++c) {
            const int kk = 4 * c + khalf;
            v2f b;
            b.x = sW2[kk * DMODEL + n * 16 + q];
            b.y = sW2[(kk + 1) * DMODEL + n * 16 + q];
            o = __builtin_amdgcn_wmma_f32_16x16x4_f32(
                false, az[c], false, b, (short)0, o, false, false);
        }
        const int d = n * 16 + q;
        const float bias2 = b2[d];
        #pragma unroll
        for (int v = 0; v < 8; ++v) {
            const int t = v + ((lane >> 4) << 3);
            out[(size_t)(tileBase + t) * DMODEL + d] = o[v] + bias2;
        }
    }
}

extern "C" void kernel_launch(void* const* d_in, const int* in_sizes, int n_in,
                              void* d_out, int out_size, void* d_ws, size_t ws_size,
                              hipStream_t stream) {
    const float* x     = (const float*)d_in[0];
    const float* W1    = (const float*)d_in[1];
    const float* b1    = (const float*)d_in[2];
    const float* theta = (const float*)d_in[3];
    const float* W2    = (const float*)d_in[4];
    const float* b2    = (const float*)d_in[5];
    float* out = (float*)d_out;

    dim3 grid(TOKENS / TOKENS_PER_BLOCK);   // 256 blocks
    dim3 block(256);                        // 8 waves (wave32)
    size_t smem = SMEM_FLOATS * sizeof(float);  // 136 KB dynamic LDS
    ffq_fused_wmma<<<grid, block, smem, stream>>>(x, W1, b1, theta, W2, b2, out);
}